// GAT_75127567942078
// MI455X (gfx1250) — compile-verified
//
#include <hip/hip_runtime.h>
#include <math.h>

// ---------------- problem constants (match reference) ----------------
static constexpr int N_NODES = 50000;
static constexpr int E_EDGES = 800000;
static constexpr int D_HID   = 64;
static constexpr int C_CLS   = 32;
static constexpr int H0 = 4, H1 = 4, H2 = 6;
static constexpr float SLOPE = 0.2f;
static constexpr int KDIM = 256;   // K is 256 for every GEMM in this net

typedef __attribute__((ext_vector_type(2))) float v2f;
typedef __attribute__((ext_vector_type(8))) float v8f;

// ------------- ordered-uint mapping for float atomic max -------------
__device__ __forceinline__ unsigned f2ord(float f) {
    unsigned u = __float_as_uint(f);
    return (u & 0x80000000u) ? ~u : (u | 0x80000000u);
}
__device__ __forceinline__ float ord2f(unsigned k) {
    return (k & 0x80000000u) ? __uint_as_float(k & 0x7fffffffu)
                             : __uint_as_float(~k);
}
// f2ord(-inf) == 0x007FFFFFu
static constexpr unsigned ORD_NEG_INF = 0x007FFFFFu;

// ---------------- weight transpose: Wt[M,K] = W[K,M]^T ----------------
__global__ void transpose_w(const float* __restrict__ W, float* __restrict__ Wt,
                            int K, int M)
{
    int idx = blockIdx.x * blockDim.x + threadIdx.x;   // over K*M
    if (idx >= K * M) return;
    int k = idx / M, m = idx % M;                      // coalesced read along m
    Wt[(size_t)m * K + k] = W[idx];
}

// =====================================================================
// WMMA fp32 GEMM:  Y[N,M] = X[N,KDIM] @ W[KDIM,M], W given TRANSPOSED
// (Wt[M,KDIM]).  1 wave per block; each wave computes a 16x64 panel.
// grid = (N/16, M/64). Requires N%16==0, M%64==0.
// A frag (16x4 f32): lane holds X[l%16][kk+2*(l/16)+{0,1}]   (one b64)
// B frag (4x16 f32): lane holds Wt[col][kk+2*(l/16)+{0,1}]   (one b64)
// D (16x16 f32): VGPR r -> row r+8*(l/16), col l%16
// Software-pipelined: next K-step's 5 fragments are prefetched before
// the current step's 4 WMMAs issue. All loads use immediate offsets
// off two base pointers (A, B) -- 16*KDIM floats = 16KB < 2^23.
// =====================================================================
__global__ __launch_bounds__(32)
void wmma_gemm_f32(const float* __restrict__ X, const float* __restrict__ Wt,
                   float* __restrict__ Y, int M)
{
    const int lane = threadIdx.x;
    const int half = lane >> 4;      // 0..1
    const int l16  = lane & 15;      // 0..15
    const int row  = blockIdx.x * 16 + l16;
    const int col  = blockIdx.y * 64 + l16;

    v8f acc0 = {}, acc1 = {}, acc2 = {}, acc3 = {};
    const float* __restrict__ Ap = X  + (size_t)row * KDIM + 2 * half;
    const float* __restrict__ Bp = Wt + (size_t)col * KDIM + 2 * half;

    // prologue: load K-step 0 fragments
    v2f a  = *(const v2f*)(Ap);
    v2f b0 = *(const v2f*)(Bp);
    v2f b1 = *(const v2f*)(Bp + 16 * KDIM);
    v2f b2 = *(const v2f*)(Bp + 32 * KDIM);
    v2f b3 = *(const v2f*)(Bp + 48 * KDIM);

    #pragma unroll 4
    for (int kk = 4; kk < KDIM; kk += 4) {
        // prefetch next K-step (5 x b64, immediate offsets)
        v2f an  = *(const v2f*)(Ap + kk);
        v2f bn0 = *(const v2f*)(Bp + kk);
        v2f bn1 = *(const v2f*)(Bp + 16 * KDIM + kk);
        v2f bn2 = *(const v2f*)(Bp + 32 * KDIM + kk);
        v2f bn3 = *(const v2f*)(Bp + 48 * KDIM + kk);
        // consume current fragments
        acc0 = __builtin_amdgcn_wmma_f32_16x16x4_f32(false, a, false, b0, (short)0, acc0, false, false);
        acc1 = __builtin_amdgcn_wmma_f32_16x16x4_f32(false, a, false, b1, (short)0, acc1, false, false);
        acc2 = __builtin_amdgcn_wmma_f32_16x16x4_f32(false, a, false, b2, (short)0, acc2, false, false);
        acc3 = __builtin_amdgcn_wmma_f32_16x16x4_f32(false, a, false, b3, (short)0, acc3, false, false);
        a = an; b0 = bn0; b1 = bn1; b2 = bn2; b3 = bn3;
    }
    // epilogue K-step
    acc0 = __builtin_amdgcn_wmma_f32_16x16x4_f32(false, a, false, b0, (short)0, acc0, false, false);
    acc1 = __builtin_amdgcn_wmma_f32_16x16x4_f32(false, a, false, b1, (short)0, acc1, false, false);
    acc2 = __builtin_amdgcn_wmma_f32_16x16x4_f32(false, a, false, b2, (short)0, acc2, false, false);
    acc3 = __builtin_amdgcn_wmma_f32_16x16x4_f32(false, a, false, b3, (short)0, acc3, false, false);

    #pragma unroll
    for (int r = 0; r < 8; ++r) {
        const int orow = blockIdx.x * 16 + r + 8 * half;
        float* __restrict__ Yp = Y + (size_t)orow * M + blockIdx.y * 64 + l16;
        Yp[0]  = acc0[r];
        Yp[16] = acc1[r];
        Yp[32] = acc2[r];
        Yp[48] = acc3[r];
    }
}

// ---------------- el/er = sum_d ft[n,h,d]*al/ar[h,d] ----------------
__global__ void dots_kernel(const float* __restrict__ ft,
                            const float* __restrict__ al,
                            const float* __restrict__ ar,
                            float* __restrict__ el, float* __restrict__ er,
                            int n_nodes, int h, int d)
{
    int idx = blockIdx.x * blockDim.x + threadIdx.x;
    if (idx >= n_nodes * h) return;
    int n = idx / h, hh = idx % h;
    const float* __restrict__ f = ft + (size_t)n * h * d + hh * d;
    const float* __restrict__ A = al + hh * d;
    const float* __restrict__ B = ar + hh * d;
    float sl = 0.f, sr = 0.f;
    for (int j = 0; j < d; j += 4) {
        float4 fv = *(const float4*)(f + j);
        float4 av = *(const float4*)(A + j);
        float4 bv = *(const float4*)(B + j);
        sl += fv.x*av.x + fv.y*av.y + fv.z*av.z + fv.w*av.w;
        sr += fv.x*bv.x + fv.y*bv.y + fv.z*bv.z + fv.w*bv.w;
    }
    el[idx] = sl; er[idx] = sr;
}

// ---------------- init emax=-inf (mapped) and denom=0 ----------------
__global__ void init_attn(unsigned* __restrict__ emaxU, float* __restrict__ denom, int count)
{
    int i = blockIdx.x * blockDim.x + threadIdx.x;
    if (i < count) { emaxU[i] = ORD_NEG_INF; denom[i] = 0.f; }
}

// acc init. mode 0: acc=bias ; mode 1: acc=bias+res ; mode 2: acc+=bias
__global__ void init_acc(float* __restrict__ acc, const float* __restrict__ bias,
                         const float* __restrict__ res, int total, int M, int mode)
{
    int i = blockIdx.x * blockDim.x + threadIdx.x;
    if (i >= total) return;
    float v = bias[i % M];
    if (mode == 1)      v += res[i];
    else if (mode == 2) v += acc[i];
    acc[i] = v;
}

// -------- pass 1: e = leaky_relu(el[src]+er[dst]); segment max -------
__global__ void edge_pass1(const int* __restrict__ src, const int* __restrict__ dst,
                           const float* __restrict__ el, const float* __restrict__ er,
                           float* __restrict__ ebuf, unsigned* __restrict__ emaxU,
                           int n_edges, int h)
{
    int e = blockIdx.x * blockDim.x + threadIdx.x;
    if (e >= n_edges) return;
    int s = src[e], dd = dst[e];
    for (int hh = 0; hh < h; ++hh) {
        float v = el[s * h + hh] + er[dd * h + hh];
        v = (v > 0.f) ? v : SLOPE * v;
        ebuf[(size_t)e * h + hh] = v;
        atomicMax(&emaxU[dd * h + hh], f2ord(v));
    }
}

// -------- pass 2: ee = exp(e - max[dst]); segment sum (denom) --------
__global__ void edge_pass2(const int* __restrict__ dst,
                           float* __restrict__ ebuf,
                           const unsigned* __restrict__ emaxU,
                           float* __restrict__ denom, int n_edges, int h)
{
    int e = blockIdx.x * blockDim.x + threadIdx.x;
    if (e >= n_edges) return;
    int dd = dst[e];
    for (int hh = 0; hh < h; ++hh) {
        float m  = ord2f(emaxU[dd * h + hh]);
        float ee = __expf(ebuf[(size_t)e * h + hh] - m);
        ebuf[(size_t)e * h + hh] = ee;
        atomicAdd(&denom[dd * h + hh], ee);
    }
}

// ---- pass 3: acc[dst,h,:] += ft[src,h,:] * (ee/denom[dst,h]) --------
// one thread per (edge, head, d/4 chunk); float4 gather + 4 f32 atomics
__global__ void edge_pass3(const int* __restrict__ src, const int* __restrict__ dst,
                           const float* __restrict__ ebuf, const float* __restrict__ denom,
                           const float* __restrict__ ft, float* __restrict__ acc,
                           int total, int h, int d4, int M)
{
    int idx = blockIdx.x * blockDim.x + threadIdx.x;
    if (idx >= total) return;
    int q  = idx % d4;
    int t  = idx / d4;
    int hh = t % h;
    int e  = t / h;
    int s = src[e], dd = dst[e];
    float alpha = ebuf[(size_t)e * h + hh] / denom[dd * h + hh];
    const float4 fv = *(const float4*)(ft + (size_t)s * M + hh * (d4 * 4) + q * 4);
    float* __restrict__ ap = acc + (size_t)dd * M + hh * (d4 * 4) + q * 4;
    atomicAdd(ap + 0, fv.x * alpha);
    atomicAdd(ap + 1, fv.y * alpha);
    atomicAdd(ap + 2, fv.z * alpha);
    atomicAdd(ap + 3, fv.w * alpha);
}

// ---------------- ELU in place ----------------
__global__ void elu_inplace(float* __restrict__ acc, int total)
{
    int i = blockIdx.x * blockDim.x + threadIdx.x;
    if (i >= total) return;
    float v = acc[i];
    acc[i] = (v > 0.f) ? v : expm1f(v);
}

// ---------------- final head-mean: [N,h,C] -> [N,C] ----------------
__global__ void finalize_mean(const float* __restrict__ acc, float* __restrict__ out,
                              int n_nodes, int h, int c)
{
    int idx = blockIdx.x * blockDim.x + threadIdx.x;
    if (idx >= n_nodes * c) return;
    int n = idx / c, cc = idx % c;
    float s = 0.f;
    for (int hh = 0; hh < h; ++hh)
        s += acc[(size_t)n * h * c + hh * c + cc];
    out[idx] = s * (1.f / h);
}

// =====================================================================
static inline int cdiv(int a, int b) { return (a + b - 1) / b; }

struct LayerBufs {
    float *el, *er, *denom, *ebuf;
    unsigned *emaxU;
};

// Wt / resWt are pre-transposed [M,K]
static void run_layer(const float* h_in, const float* Wt, const float* al, const float* ar,
                      const float* bias, const float* resWt, int resMode, // 0 none,1 identity,2 gemm
                      float* ft, float* acc, const int* src, const int* dst,
                      int h, int d, bool do_elu, const LayerBufs& B, hipStream_t stream)
{
    const int M = h * d;

    // 1) ft = h_in @ W   (WMMA fp32, B from transposed weights)
    dim3 ggrid(N_NODES / 16, M / 64);
    wmma_gemm_f32<<<ggrid, 32, 0, stream>>>(h_in, Wt, ft, M);

    // 2) accumulator init (bias + residual)
    const int totNM = N_NODES * M;
    if (resMode == 2) {
        wmma_gemm_f32<<<ggrid, 32, 0, stream>>>(h_in, resWt, acc, M);
        init_acc<<<cdiv(totNM, 256), 256, 0, stream>>>(acc, bias, nullptr, totNM, M, 2);
    } else if (resMode == 1) {
        init_acc<<<cdiv(totNM, 256), 256, 0, stream>>>(acc, bias, h_in, totNM, M, 1);
    } else {
        init_acc<<<cdiv(totNM, 256), 256, 0, stream>>>(acc, bias, nullptr, totNM, M, 0);
    }

    // 3) attention dot products
    dots_kernel<<<cdiv(N_NODES * h, 256), 256, 0, stream>>>(ft, al, ar, B.el, B.er,
                                                            N_NODES, h, d);
    // 4) segment max / sum scratch init
    init_attn<<<cdiv(N_NODES * h, 256), 256, 0, stream>>>(B.emaxU, B.denom, N_NODES * h);

    // 5-7) edge softmax + aggregation
    edge_pass1<<<cdiv(E_EDGES, 256), 256, 0, stream>>>(src, dst, B.el, B.er, B.ebuf,
                                                       B.emaxU, E_EDGES, h);
    edge_pass2<<<cdiv(E_EDGES, 256), 256, 0, stream>>>(dst, B.ebuf, B.emaxU, B.denom,
                                                       E_EDGES, h);
    const int d4 = d / 4;
    const int tot3 = E_EDGES * h * d4;
    edge_pass3<<<cdiv(tot3, 256), 256, 0, stream>>>(src, dst, B.ebuf, B.denom, ft, acc,
                                                    tot3, h, d4, M);
    // 8) activation
    if (do_elu)
        elu_inplace<<<cdiv(totNM, 256), 256, 0, stream>>>(acc, totNM);
}

extern "C" void kernel_launch(void* const* d_in, const int* in_sizes, int n_in,
                              void* d_out, int out_size, void* d_ws, size_t ws_size,
                              hipStream_t stream)
{
    (void)in_sizes; (void)n_in; (void)out_size; (void)ws_size;

    const float* x    = (const float*)d_in[0];
    const int*   src  = (const int*)d_in[1];
    const int*   dst  = (const int*)d_in[2];
    const float* W0   = (const float*)d_in[3];
    const float* al0  = (const float*)d_in[4];
    const float* ar0  = (const float*)d_in[5];
    const float* b0   = (const float*)d_in[6];
    const float* W1   = (const float*)d_in[7];
    const float* al1  = (const float*)d_in[8];
    const float* ar1  = (const float*)d_in[9];
    const float* b1   = (const float*)d_in[10];
    const float* W2   = (const float*)d_in[11];
    const float* al2  = (const float*)d_in[12];
    const float* ar2  = (const float*)d_in[13];
    const float* b2   = (const float*)d_in[14];
    const float* rW2  = (const float*)d_in[15];
    float* out = (float*)d_out;

    // ---- workspace carve-up (floats) ----
    const size_t NB = (size_t)N_NODES * 256;   // big node buffer
    float* f = (float*)d_ws;
    float* bufA = f;            f += NB;       // ft scratch (all layers)
    float* bufB = f;            f += NB;       // layer0 out / layer2 acc
    float* bufC = f;            f += NB;       // layer1 out
    float* el    = f;           f += (size_t)N_NODES * 8;
    float* er    = f;           f += (size_t)N_NODES * 8;
    float* denom = f;           f += (size_t)N_NODES * 8;
    unsigned* emaxU = (unsigned*)f; f += (size_t)N_NODES * 8;
    float* ebuf  = f;           f += (size_t)E_EDGES * 6;
    float* wt0   = f;           f += (size_t)KDIM * 256;   // W0^T  [256,256]
    float* wt1   = f;           f += (size_t)KDIM * 256;   // W1^T  [256,256]
    float* wt2   = f;           f += (size_t)KDIM * 192;   // W2^T  [192,256]
    float* wtR   = f;           f += (size_t)KDIM * 192;   // resW2^T

    LayerBufs B{el, er, denom, ebuf, emaxU};

    // ---- transpose all weights once (tiny: <= 256 KB each) ----
    transpose_w<<<cdiv(KDIM * 256, 256), 256, 0, stream>>>(W0,  wt0, KDIM, 256);
    transpose_w<<<cdiv(KDIM * 256, 256), 256, 0, stream>>>(W1,  wt1, KDIM, 256);
    transpose_w<<<cdiv(KDIM * 192, 256), 256, 0, stream>>>(W2,  wt2, KDIM, 192);
    transpose_w<<<cdiv(KDIM * 192, 256), 256, 0, stream>>>(rW2, wtR, KDIM, 192);

    // layer 0: x[N,256] -> bufB [N, 4*64], ELU
    run_layer(x,    wt0, al0, ar0, b0, nullptr, /*res*/0, bufA, bufB, src, dst,
              H0, D_HID, /*elu*/true,  B, stream);
    // layer 1: bufB -> bufC, identity residual, ELU
    run_layer(bufB, wt1, al1, ar1, b1, nullptr, /*res*/1, bufA, bufC, src, dst,
              H1, D_HID, /*elu*/true,  B, stream);
    // layer 2: bufC -> bufB [N, 6*32], GEMM residual, no act
    run_layer(bufC, wt2, al2, ar2, b2, wtR,    /*res*/2, bufA, bufB, src, dst,
              H2, C_CLS, /*elu*/false, B, stream);

    // mean over heads -> d_out [N, 32]
    finalize_mean<<<cdiv(N_NODES * C_CLS, 256), 256, 0, stream>>>(bufB, out,
                                                                  N_NODES, H2, C_CLS);
}